// QuantumDNN_66400194396348
// MI455X (gfx1250) — compile-verified
//
#include <hip/hip_runtime.h>
#include <hip/hip_bf16.h>

// MI455X (gfx1250) quantum-circuit MLP.
// Each circuit layer == dense 64x64 complex unitary -> batched complex GEMM
// on V_WMMA_F32_16X16X4_F32; psi resident in LDS; per-layer U matrices
// double-buffered in LDS via GLOBAL_LOAD_ASYNC_TO_LDS_B128 (ASYNCcnt).

typedef __attribute__((ext_vector_type(2))) float v2f;
typedef __attribute__((ext_vector_type(8))) float v8f;

#define NQ 6
#define DIM 64            // 2^NQ amplitudes
#define NLAYERS 64        // 8 blocks * 8 layers
#define B_TOTAL 8192
#define PSTRIDE 68        // psi LDS row stride (floats), padded vs 64 banks
#define USTRIDE 68        // U   LDS row stride (floats)
#define WAVES 4
#define THREADS (WAVES * 32)
#define CHUNK (WAVES * 16)            // batch rows per workgroup
#define NWG (B_TOTAL / CHUNK)

// ---------------------------------------------------------------------------
// Kernel 1: build per-layer transposed unitaries Ut = U^T (real & imag parts)
// U[sigma(b)][j] = prod_w R_w[b_w][j_w]; sigma = CNOT ring permutation.
// Layout in d_ws: layer L -> [Br(64x64) | Bi(64x64)] row-major, row=j, col=i.
// ---------------------------------------------------------------------------
__global__ void build_U_kernel(const float* __restrict__ qw,
                               float* __restrict__ Umats) {
  const int L = blockIdx.x;          // global layer 0..63
  const int l = L & 7;               // layer within block
  const int r = (l % (NQ - 1)) + 1;  // CNOT range

  __shared__ float Rre[NQ][4];
  __shared__ float Rim[NQ][4];
  const int tid = threadIdx.x;

  if (tid < NQ) {
    const float* q = qw + ((size_t)L * NQ + tid) * 3;
    const float phi = q[0], th = q[1], om = q[2];
    float s, c;   sincosf(0.5f * th, &s, &c);
    float sp, cp; sincosf(0.5f * (phi + om), &sp, &cp);  // ep = (cp, -sp)
    float sm, cm; sincosf(0.5f * (phi - om), &sm, &cm);  // em = (cm,  sm)
    Rre[tid][0] =  c * cp;  Rim[tid][0] = -c * sp;   // R00 =  c*ep
    Rre[tid][1] = -s * cm;  Rim[tid][1] = -s * sm;   // R01 = -s*em
    Rre[tid][2] =  s * cm;  Rim[tid][2] = -s * sm;   // R10 =  s*conj(em)
    Rre[tid][3] =  c * cp;  Rim[tid][3] =  c * sp;   // R11 =  c*conj(ep)
  }
  __syncthreads();

  float* base = Umats + (size_t)L * 2 * DIM * DIM;
  for (int id = tid; id < DIM * DIM; id += blockDim.x) {
    const int b = id >> 6;
    const int j = id & 63;
    // sigma(b): apply CNOT(i, (i+r)%6) for i = 0..5 in order.
    int bp = b;
#pragma unroll
    for (int i = 0; i < NQ; ++i) {
      int t = i + r; if (t >= NQ) t -= NQ;
      if ((bp >> (5 - i)) & 1) bp ^= (1 << (5 - t));
    }
    // Kron[b][j] = prod_w R_w[b_w][j_w] (complex)
    float pr = 1.f, pi = 0.f;
#pragma unroll
    for (int w = 0; w < NQ; ++w) {
      const int ib = (b >> (5 - w)) & 1;
      const int jb = (j >> (5 - w)) & 1;
      const float rr = Rre[w][ib * 2 + jb];
      const float ri = Rim[w][ib * 2 + jb];
      const float nr = pr * rr - pi * ri;
      pi = pr * ri + pi * rr;
      pr = nr;
    }
    base[0 * DIM * DIM + j * DIM + bp] = pr;  // Br[j][sigma(b)]
    base[1 * DIM * DIM + j * DIM + bp] = pi;  // Bi[j][sigma(b)]
  }
}

// ---------------------------------------------------------------------------
// Async global->LDS staging of one layer's Ut pair (32 KB) with row padding.
// b128 per lane, tracked by ASYNCcnt; no VGPR data movement.
// ---------------------------------------------------------------------------
__device__ __forceinline__ void stage_async(const float* __restrict__ Umats,
                                            int L, float* dstbase, int tid) {
  const float* src = Umats + (size_t)L * 2 * DIM * DIM;
#pragma unroll 4
  for (int e = tid; e < 2 * DIM * DIM / 4; e += THREADS) {
    const int elt = e << 2;
    const int m  = elt >> 12;       // 0 = real, 1 = imag
    const int rc = elt & 4095;
    const int rr = rc >> 6;
    const int cc = rc & 63;
    const unsigned lds =
        (unsigned)(size_t)(dstbase + m * (DIM * USTRIDE) + rr * USTRIDE + cc);
    const unsigned long long ga = (unsigned long long)(size_t)(src + elt);
    asm volatile("global_load_async_to_lds_b128 %0, %1, off"
                 :: "v"(lds), "v"(ga) : "memory");
  }
}

// ---------------------------------------------------------------------------
// Kernel 2: fused embedding + 64 WMMA layers + readout + MLP tail.
// ---------------------------------------------------------------------------
__global__ void __launch_bounds__(THREADS) qsim_kernel(
    const float* __restrict__ x,  const float* __restrict__ W1,
    const float* __restrict__ b1, const float* __restrict__ W2,
    const float* __restrict__ b2, const float* __restrict__ W3,
    const float* __restrict__ b3, const float* __restrict__ Umats,
    float* __restrict__ out) {
  __shared__ __align__(16) float sRe[CHUNK * PSTRIDE];
  __shared__ __align__(16) float sIm[CHUNK * PSTRIDE];
  __shared__ __align__(16) float sU[2][2][DIM * USTRIDE];  // double buffer

  const int tid = threadIdx.x;

  // kick off async staging of layer 0 into buffer 0 before anything else
  stage_async(Umats, 0, &sU[0][0][0], tid);

  // ---- init: RX embedding. 2 threads per batch row, 32 amplitudes each ----
  {
    const int row  = tid & (CHUNK - 1);
    const int half = tid >> 6;  // CHUNK == 64
    const int grow = blockIdx.x * CHUNK + row;
    const float x0 = x[grow * 3 + 0];
    const float x1 = x[grow * 3 + 1];
    const float x2 = x[grow * 3 + 2];
    float cw[NQ], sw[NQ];
#pragma unroll
    for (int q = 0; q < NQ; ++q) {
      const float a = x0 * W1[q * 3 + 0] + x1 * W1[q * 3 + 1] +
                      x2 * W1[q * 3 + 2] + b1[q];
      sincosf(0.5f * a, &sw[q], &cw[q]);
    }
    for (int idx = half * 32; idx < half * 32 + 32; ++idx) {
      float mag = 1.f;
#pragma unroll
      for (int w = 0; w < NQ; ++w)
        mag *= ((idx >> (5 - w)) & 1) ? sw[w] : cw[w];
      const int k2 = __popc(idx) & 3;  // amplitude phase = (-i)^popcount
      float re = 0.f, im = 0.f;
      if      (k2 == 0) re =  mag;
      else if (k2 == 1) im = -mag;
      else if (k2 == 2) re = -mag;
      else              im =  mag;
      sRe[row * PSTRIDE + idx] = re;
      sIm[row * PSTRIDE + idx] = im;
    }
  }

  const int wave  = tid >> 5;
  const int lane  = tid & 31;
  const int laneM = lane & 15;            // A: row M / B: col N / C: col N
  const int koff  = (lane >> 4) << 1;     // K sub-offset 0 or 2
  const int m0    = wave * 16;            // this wave's psi strip base row
  const int rwb   = m0 + ((lane >> 4) << 3);  // C/D writeback row base

  for (int L = 0; L < NLAYERS; ++L) {
    const int buf = L & 1;
    // our async loads into `buf` have landed; barrier publishes them to all
    // waves AND proves every wave is done reading buf^1 (layer L-1).
    asm volatile("s_wait_asynccnt 0x0" ::: "memory");
    __syncthreads();

    // prefetch next layer into the buffer everyone just finished with
    if (L + 1 < NLAYERS)
      stage_async(Umats, L + 1, &sU[buf ^ 1][0][0], tid);

    // complex GEMM: Y[16x64] = X[16x64] * Ut, via f32 WMMA 16x16x4
    v8f accR[4], accI[4];
#pragma unroll
    for (int n = 0; n < 4; ++n) {
      accR[n] = (v8f){0, 0, 0, 0, 0, 0, 0, 0};
      accI[n] = (v8f){0, 0, 0, 0, 0, 0, 0, 0};
    }
    const float* pre = &sRe[(m0 + laneM) * PSTRIDE];
    const float* pim = &sIm[(m0 + laneM) * PSTRIDE];
    const float* Ur  = sU[buf][0];
    const float* Ui  = sU[buf][1];

    for (int kk = 0; kk < 16; ++kk) {
      const int k0 = kk * 4 + koff;
      v2f aRe, aIm, aImN;
      aRe.x = pre[k0];  aRe.y = pre[k0 + 1];
      aIm.x = pim[k0];  aIm.y = pim[k0 + 1];
      aImN.x = -aIm.x;  aImN.y = -aIm.y;   // f32 WMMA NEG is C-only (ISA)
      const int r0 = k0 * USTRIDE + laneM;
      const int r1 = r0 + USTRIDE;
#pragma unroll
      for (int nt = 0; nt < 4; ++nt) {
        const int cb = nt * 16;
        v2f bR, bI;
        bR.x = Ur[r0 + cb];  bR.y = Ur[r1 + cb];
        bI.x = Ui[r0 + cb];  bI.y = Ui[r1 + cb];
        // Yr += Xr*Br - Xi*Bi ; Yi += Xr*Bi + Xi*Br
        accR[nt] = __builtin_amdgcn_wmma_f32_16x16x4_f32(
            false, aRe,  false, bR, (short)0, accR[nt], false, false);
        accR[nt] = __builtin_amdgcn_wmma_f32_16x16x4_f32(
            false, aImN, false, bI, (short)0, accR[nt], false, false);
        accI[nt] = __builtin_amdgcn_wmma_f32_16x16x4_f32(
            false, aRe,  false, bI, (short)0, accI[nt], false, false);
        accI[nt] = __builtin_amdgcn_wmma_f32_16x16x4_f32(
            false, aIm,  false, bR, (short)0, accI[nt], false, false);
      }
    }

    // write back C-layout -> psi LDS (wave-private strip, DS in-order)
#pragma unroll
    for (int nt = 0; nt < 4; ++nt) {
      const int col = nt * 16 + laneM;
#pragma unroll
      for (int v = 0; v < 8; ++v) {
        sRe[(rwb + v) * PSTRIDE + col] = accR[nt][v];
        sIm[(rwb + v) * PSTRIDE + col] = accI[nt][v];
      }
    }
  }

  __syncthreads();

  // ---- readout <Z_q> + MLP tail, one thread per batch row ----
  if (tid < CHUNK) {
    float z[NQ] = {0, 0, 0, 0, 0, 0};
    for (int idx = 0; idx < DIM; ++idx) {
      const float re = sRe[tid * PSTRIDE + idx];
      const float im = sIm[tid * PSTRIDE + idx];
      const float p = re * re + im * im;
#pragma unroll
      for (int q = 0; q < NQ; ++q)
        z[q] += ((idx >> (5 - q)) & 1) ? -p : p;
    }
    float h[64];
    for (int j = 0; j < 64; ++j) {
      float s = b2[j];
#pragma unroll
      for (int q = 0; q < NQ; ++q) s += W2[j * NQ + q] * z[q];
      h[j] = tanhf(s);
    }
    const int gr = blockIdx.x * CHUNK + tid;
#pragma unroll
    for (int o = 0; o < 4; ++o) {
      float s = b3[o];
      for (int j = 0; j < 64; ++j) s += W3[o * 64 + j] * h[j];
      out[gr * 4 + o] = s;
    }
  }
}

extern "C" void kernel_launch(void* const* d_in, const int* in_sizes, int n_in,
                              void* d_out, int out_size, void* d_ws,
                              size_t ws_size, hipStream_t stream) {
  (void)in_sizes; (void)n_in; (void)out_size; (void)ws_size;
  const float* x  = (const float*)d_in[0];
  const float* W1 = (const float*)d_in[1];
  const float* b1 = (const float*)d_in[2];
  const float* qw = (const float*)d_in[3];
  const float* W2 = (const float*)d_in[4];
  const float* b2 = (const float*)d_in[5];
  const float* W3 = (const float*)d_in[6];
  const float* b3 = (const float*)d_in[7];
  float* Umats = (float*)d_ws;  // 64 layers * 2 * 64*64 f32 = 2 MB

  build_U_kernel<<<NLAYERS, 256, 0, stream>>>(qw, Umats);
  qsim_kernel<<<NWG, THREADS, 0, stream>>>(x, W1, b1, W2, b2, W3, b3, Umats,
                                           (float*)d_out);
}